// HybridDecompConv_43173011259747
// MI455X (gfx1250) — compile-verified
//
#include <hip/hip_runtime.h>
#include <math.h>

// Problem constants (from reference)
#define BB    16
#define CCH   128
#define LLEN  4096
#define CO    256
#define KDIM  512          // 4 bands * C
#define MTOT  65536        // B * L

typedef float v2f __attribute__((ext_vector_type(2)));
typedef float v8f __attribute__((ext_vector_type(8)));

// ---------------------------------------------------------------------------
// Kernel 1: depthwise conv (k=3,5,7,11) + BN(eval) + exact GELU + scale.
// One (b,c) row per 256-thread block; row cached in LDS with zero halo.
// Writes alpha_s * softmax_i * gelu(...) into catT[(i*C+c), b*L + l]  (K-major
// cat layout -> fully coalesced stores along l).
// ---------------------------------------------------------------------------
__global__ __launch_bounds__(256) void conv_bn_gelu_kernel(
    const float* __restrict__ x,
    const float* __restrict__ w0, const float* __restrict__ w1,
    const float* __restrict__ w2, const float* __restrict__ w3,
    const float* __restrict__ gamma, const float* __restrict__ beta,
    const float* __restrict__ mean,  const float* __restrict__ var,
    const float* __restrict__ sw,    const float* __restrict__ alpha,
    float* __restrict__ catT)
{
    __shared__ float xrow[LLEN + 16];   // 8-wide zero halo each side
    const int tid = threadIdx.x;
    const int c   = blockIdx.x & (CCH - 1);
    const int b   = blockIdx.x >> 7;

    const float* xr = x + (size_t)(b * CCH + c) * LLEN;
    #pragma unroll
    for (int r = 0; r < 16; ++r) xrow[8 + tid + r * 256] = xr[tid + r * 256];
    if (tid < 8) { xrow[tid] = 0.f; xrow[8 + LLEN + tid] = 0.f; }
    __syncthreads();

    // softmax over the 4 scale weights + sigmoid(alpha) (tiny, recompute/thread)
    float s0 = sw[0], s1 = sw[1], s2 = sw[2], s3 = sw[3];
    float mx = fmaxf(fmaxf(s0, s1), fmaxf(s2, s3));
    float e0 = expf(s0 - mx), e1 = expf(s1 - mx), e2 = expf(s2 - mx), e3 = expf(s3 - mx);
    float einv = 1.f / (e0 + e1 + e2 + e3);
    float nwv[4] = {e0 * einv, e1 * einv, e2 * einv, e3 * einv};
    float alpha_s = 1.f / (1.f + expf(-alpha[0]));

    const float* wts[4] = {w0, w1, w2, w3};
    const int ks[4] = {3, 5, 7, 11};

    #pragma unroll
    for (int band = 0; band < 4; ++band) {
        const int k = ks[band];
        const int half = k >> 1;
        float wreg[11];
        #pragma unroll
        for (int j = 0; j < 11; ++j) wreg[j] = (j < k) ? wts[band][c * k + j] : 0.f;

        const float mu  = mean[band * CCH + c];
        const float inv = rsqrtf(var[band * CCH + c] + 1e-5f) * gamma[band * CCH + c];
        const float bta = beta[band * CCH + c];
        const float sc  = alpha_s * nwv[band];

        float* dst = catT + (size_t)(band * CCH + c) * MTOT + (size_t)b * LLEN;
        #pragma unroll
        for (int r = 0; r < 16; ++r) {
            int l = tid + r * 256;
            float acc = 0.f;
            for (int j = 0; j < k; ++j) acc += xrow[8 + l - half + j] * wreg[j];
            float y  = (acc - mu) * inv + bta;
            float ge = 0.5f * y * (1.f + erff(y * 0.70710678118654752f)); // exact GELU
            dst[l] = sc * ge;
        }
    }
}

// ---------------------------------------------------------------------------
// Stockham radix-2 DIF FFT, N=4096, 256 threads, ping-pong LDS buffers.
// 12 stages (even) -> result lands back in (pr, pi). sign=-1 fwd, +1 inv.
// ---------------------------------------------------------------------------
__device__ __forceinline__ void fft4096(float* pr, float* pi_, float* qr, float* qi,
                                        float sign, int tid)
{
    int n = 4096, ls = 0;
    #pragma unroll 1
    for (int stage = 0; stage < 12; ++stage) {
        const int   half  = n >> 1;
        const float theta = sign * 6.2831853071795864769f / (float)n;
        const int   s     = 1 << ls;
        #pragma unroll
        for (int u = 0; u < 8; ++u) {
            int t = tid + u * 256;              // 0..2047 butterflies
            int p = t >> ls;
            int q = t & (s - 1);
            float sn, cs;
            __sincosf(theta * (float)p, &sn, &cs);
            int ia = q + (p << ls);
            int ib = ia + (half << ls);
            float ar = pr[ia], ai = pi_[ia];
            float br = pr[ib], bi = pi_[ib];
            int io = q + (p << (ls + 1));
            qr[io] = ar + br;
            qi[io] = ai + bi;
            float dr = ar - br, di = ai - bi;
            qr[io + s] = dr * cs - di * sn;
            qi[io + s] = dr * sn + di * cs;
        }
        __syncthreads();
        n >>= 1; ls += 1;
        float* t0;
        t0 = pr;  pr  = qr; qr = t0;
        t0 = pi_; pi_ = qi; qi = t0;
    }
}

// ---------------------------------------------------------------------------
// Kernel 2: frequency-band branch. Packs TWO real channels into one complex
// FFT (Z = FFT(x_c0 + i x_c1)); band masks are Hermitian-symmetric so
// IFFT(M.Z) = band_c0 + i band_c1 directly. Accumulates (1-alpha_s)/N * band
// into catT (RMW, coalesced).
// Dynamic LDS: work re/im + scratch re/im + retained spectrum = 96 KB.
// ---------------------------------------------------------------------------
__global__ __launch_bounds__(256) void fft_band_kernel(
    const float* __restrict__ x, const float* __restrict__ alpha,
    float* __restrict__ catT)
{
    extern __shared__ float fsm[];
    float* wr = fsm;
    float* wi = fsm + 4096;
    float* yr = fsm + 8192;
    float* yi = fsm + 12288;
    float* sr = fsm + 16384;
    float* si = fsm + 20480;

    const int tid = threadIdx.x;
    const int g   = blockIdx.x & 63;
    const int b   = blockIdx.x >> 6;
    const int c0  = 2 * g, c1 = 2 * g + 1;

    const float* x0 = x + (size_t)(b * CCH + c0) * LLEN;
    const float* x1 = x + (size_t)(b * CCH + c1) * LLEN;
    #pragma unroll
    for (int u = 0; u < 16; ++u) {
        int l = tid + u * 256;
        wr[l] = x0[l];
        wi[l] = x1[l];
    }
    __syncthreads();

    fft4096(wr, wi, yr, yi, -1.f, tid);                 // forward

    #pragma unroll
    for (int u = 0; u < 16; ++u) { int f = tid + u * 256; sr[f] = wr[f]; si[f] = wi[f]; }
    float alpha_s = 1.f / (1.f + expf(-alpha[0]));
    const float oscale = (1.f - alpha_s) * (1.f / 4096.f);
    __syncthreads();

    for (int band = 0; band < 4; ++band) {
        // band_size = 2049/4 = 512; bands [0,512),[512,1024),[1024,1536),[1536,2049)
        #pragma unroll
        for (int u = 0; u < 16; ++u) {
            int f  = tid + u * 256;
            int gg = (f <= 2048) ? f : (4096 - f);      // mirrored bin
            int bi = gg >> 9; bi = bi > 3 ? 3 : bi;     // Nyquist -> band 3
            bool keep = (bi == band);
            wr[f] = keep ? sr[f] : 0.f;
            wi[f] = keep ? si[f] : 0.f;
        }
        __syncthreads();
        fft4096(wr, wi, yr, yi, 1.f, tid);              // inverse (unscaled)

        float* d0 = catT + (size_t)(band * CCH + c0) * MTOT + (size_t)b * LLEN;
        float* d1 = catT + (size_t)(band * CCH + c1) * MTOT + (size_t)b * LLEN;
        #pragma unroll
        for (int u = 0; u < 16; ++u) {
            int l = tid + u * 256;
            d0[l] += oscale * wr[l];
            d1[l] += oscale * wi[l];
        }
        __syncthreads();
    }
}

// ---------------------------------------------------------------------------
// Kernel 3: projection GEMM via fp32 WMMA (V_WMMA_F32_16X16X4_F32).
// out[m, n] = sum_k catT[k, m] * proj_w[n, k] + proj_b[n]
// Block = 256 threads = 8 waves in a 2(M) x 4(N) grid -> 32x256 tile.
//
// K-pair LDS layout: fragments (k, k+1) per lane are adjacent -> each WMMA
// operand is a single aligned ds_load_b64, no v_mov repacking.
//   As2[kpair][m][2]  stride 96  floats per kpair (96  mod 64 == 32)
//   Bs2[kpair][n][2]  stride 544 floats per kpair (544 mod 64 == 32)
// The mod-64==32 strides put lanes 0-15 (kpair p) and lanes 16-31 (kpair p+1)
// in disjoint LDS bank halves -> conflict-free b64 fragment loads.
// ---------------------------------------------------------------------------
#define ASTRIDE 96            // floats per kpair in As2 (32 m * 2 + 32 pad)
#define BSTRIDE 544           // floats per kpair in Bs2 (256 n * 2 + 32 pad)

__global__ __launch_bounds__(256) void proj_wmma_kernel(
    const float* __restrict__ catT, const float* __restrict__ pw,
    const float* __restrict__ pb,   float* __restrict__ out)
{
    extern __shared__ float gsm[];
    float* As2 = gsm;                   // 32 kpairs * 96  = 3072 floats
    float* Bs2 = gsm + 32 * ASTRIDE;    // 32 kpairs * 544 = 17408 floats
    float* Cs  = gsm;                   // alias after compute: [256][33]
    v2f* As2v = (v2f*)As2;
    v2f* Bs2v = (v2f*)Bs2;

    const int tid  = threadIdx.x;
    const int lane = tid & 31;
    const int wave = tid >> 5;
    const int mw   = (wave >> 2) * 16;   // 0 or 16
    const int nw   = (wave & 3) * 64;    // 0,64,128,192
    const int m0   = blockIdx.x * 32;

    const int arow = lane & 15;          // M/N column within fragment
    const int klo  = lane >> 4;          // lanes 16-31 take kpair+1 (K+2,K+3)

    v8f acc0 = {}, acc1 = {}, acc2 = {}, acc3 = {};

    for (int chunk = 0; chunk < 8; ++chunk) {
        const int k0 = chunk * 64;
        __syncthreads();
        // A tile: catT is K-major -> coalesced 32-float rows per k
        #pragma unroll
        for (int r = 0; r < 8; ++r) {
            int idx = tid + r * 256;
            int k = idx >> 5, m = idx & 31;
            As2[(k >> 1) * ASTRIDE + m * 2 + (k & 1)] =
                catT[(size_t)(k0 + k) * MTOT + m0 + m];
        }
        // B tile: pw row-major [256][512]; thread tid loads its n-row slice
        // (L2-hot weight) and stores K-pairs as b64s.
        {
            const float4* src = (const float4*)(pw + (size_t)tid * KDIM + k0);
            #pragma unroll
            for (int j = 0; j < 16; ++j) {
                float4 v = src[j];
                v2f lo; lo.x = v.x; lo.y = v.y;
                v2f hi; hi.x = v.z; hi.y = v.w;
                Bs2v[(2 * j)     * (BSTRIDE / 2) + tid] = lo;
                Bs2v[(2 * j + 1) * (BSTRIDE / 2) + tid] = hi;
            }
        }
        if (chunk < 7)
            __builtin_prefetch(catT + (size_t)(k0 + 64) * MTOT + m0, 0, 1);
        __syncthreads();

        #pragma unroll
        for (int kk = 0; kk < 64; kk += 4) {
            const int kp = (kk >> 1) + klo;     // this lane's kpair
            v2f a  = As2v[kp * (ASTRIDE / 2) + mw + arow];
            v2f b0 = Bs2v[kp * (BSTRIDE / 2) + nw + arow];
            v2f b1 = Bs2v[kp * (BSTRIDE / 2) + nw + 16 + arow];
            v2f b2 = Bs2v[kp * (BSTRIDE / 2) + nw + 32 + arow];
            v2f b3 = Bs2v[kp * (BSTRIDE / 2) + nw + 48 + arow];
            acc0 = __builtin_amdgcn_wmma_f32_16x16x4_f32(false, a, false, b0, (short)0, acc0, false, false);
            acc1 = __builtin_amdgcn_wmma_f32_16x16x4_f32(false, a, false, b1, (short)0, acc1, false, false);
            acc2 = __builtin_amdgcn_wmma_f32_16x16x4_f32(false, a, false, b2, (short)0, acc2, false, false);
            acc3 = __builtin_amdgcn_wmma_f32_16x16x4_f32(false, a, false, b3, (short)0, acc3, false, false);
        }
    }
    __syncthreads();

    // bias + stage D into Cs[n][m] (D layout: vgpr r -> M = r + 8*(lane>=16))
    {
        float pbv0 = pb[nw + arow];
        float pbv1 = pb[nw + 16 + arow];
        float pbv2 = pb[nw + 32 + arow];
        float pbv3 = pb[nw + 48 + arow];
        int mbase = mw + 8 * (lane >> 4);
        #pragma unroll
        for (int r = 0; r < 8; ++r) {
            int m = mbase + r;
            Cs[(nw + arow) * 33 + m]      = acc0[r] + pbv0;
            Cs[(nw + 16 + arow) * 33 + m] = acc1[r] + pbv1;
            Cs[(nw + 32 + arow) * 33 + m] = acc2[r] + pbv2;
            Cs[(nw + 48 + arow) * 33 + m] = acc3[r] + pbv3;
        }
    }
    __syncthreads();

    // coalesced store: thread tid owns output row n=tid, 32 contiguous l's
    {
        const int bidx = m0 >> 12;          // L = 4096, tile never crosses b
        const int l0   = m0 & 4095;
        float* dst = out + ((size_t)(bidx * CO + tid)) * LLEN + l0;
        #pragma unroll
        for (int v = 0; v < 8; ++v) {
            float4 o;
            o.x = Cs[tid * 33 + v * 4 + 0];
            o.y = Cs[tid * 33 + v * 4 + 1];
            o.z = Cs[tid * 33 + v * 4 + 2];
            o.w = Cs[tid * 33 + v * 4 + 3];
            ((float4*)dst)[v] = o;
        }
    }
}

// ---------------------------------------------------------------------------
extern "C" void kernel_launch(void* const* d_in, const int* in_sizes, int n_in,
                              void* d_out, int out_size, void* d_ws, size_t ws_size,
                              hipStream_t stream)
{
    const float* x     = (const float*)d_in[0];
    const float* w0    = (const float*)d_in[1];
    const float* w1    = (const float*)d_in[2];
    const float* w2    = (const float*)d_in[3];
    const float* w3    = (const float*)d_in[4];
    const float* gamma = (const float*)d_in[5];
    const float* beta  = (const float*)d_in[6];
    const float* mean  = (const float*)d_in[7];
    const float* var   = (const float*)d_in[8];
    const float* sw    = (const float*)d_in[9];
    const float* alpha = (const float*)d_in[10];
    const float* pw    = (const float*)d_in[11];
    const float* pb    = (const float*)d_in[12];

    float* catT = (float*)d_ws;            // [512][65536] fp32 = 128 MB
    float* outp = (float*)d_out;           // [16][256][4096] fp32

    // K1: conv branch -> writes every catT element (no init needed)
    conv_bn_gelu_kernel<<<BB * CCH, 256, 0, stream>>>(
        x, w0, w1, w2, w3, gamma, beta, mean, var, sw, alpha, catT);

    // K2: FFT band branch (2 channels per block), accumulates into catT
    fft_band_kernel<<<BB * (CCH / 2), 256, 24576 * sizeof(float), stream>>>(
        x, alpha, catT);

    // K3: WMMA projection + transpose store
    proj_wmma_kernel<<<MTOT / 32, 256,
                       (32 * ASTRIDE + 32 * BSTRIDE) * sizeof(float), stream>>>(
        catT, pw, pb, outp);
}